// GraphNeuralNetwork_46583215292429
// MI455X (gfx1250) — compile-verified
//
#include <hip/hip_runtime.h>
#include <hip/hip_bf16.h>

typedef float v2f __attribute__((ext_vector_type(2)));
typedef float v4f __attribute__((ext_vector_type(4)));
typedef float v8f __attribute__((ext_vector_type(8)));

#define BN_EPS 1e-5f

// ---------------------------------------------------------------------------
// Degree / normalization: deg[i] = in-degree + 1 (self loop), dinv = rsqrt(deg)
// ---------------------------------------------------------------------------
__global__ void gcn_deg_init(float* __restrict__ deg, int n) {
  int i = blockIdx.x * blockDim.x + threadIdx.x;
  if (i < n) deg[i] = 1.0f;  // self loop
}

__global__ void gcn_deg_accum(const int* __restrict__ dst, float* __restrict__ deg, int e) {
  int i = blockIdx.x * blockDim.x + threadIdx.x;
  if (i < e) atomicAdd(&deg[dst[i]], 1.0f);
}

__global__ void gcn_deg_to_dinv(float* __restrict__ deg, int n) {
  int i = blockIdx.x * blockDim.x + threadIdx.x;
  if (i < n) deg[i] = rsqrtf(deg[i]);
}

// ---------------------------------------------------------------------------
// C[M x NCOL] = A[M x K] @ W[K x NCOL], all row-major fp32.
// WMMA V_WMMA_F32_16X16X4_F32: one wave owns a 16x16 C tile, K-loop step 4.
// The 16xK A tile is staged in LDS once per block (all waves share the same
// rows, different column blocks) -> 8x less L0/L2 traffic than direct loads.
// Row pad of 4 floats makes the A-fragment read pattern hit 64 distinct banks.
//
// Fragment layouts per CDNA5 ISA 7.12.2 (32-bit matrices, wave32):
//   A: lane L -> row (L&15), K = {k+2*(L>>4), k+2*(L>>4)+1}  (v2f)
//   B: lane L -> col (L&15), same K pair mapping             (v2f)
//   C: VGPR r -> row r + 8*(L>>4), col (L&15)                (v8f)
// Requires M % 16 == 0 (100000/16 = 6250), K % 4 == 0, NCOL % 16 == 0.
// blockDim.x = 32 * (NCOL/16); gridDim.x = M/16.
// ---------------------------------------------------------------------------
template <int K, int NCOL>
__global__ void gcn_gemm_wmma(const float* __restrict__ A,
                              const float* __restrict__ W,
                              float* __restrict__ C) {
  constexpr int PAD = 4;                 // floats of row padding (bank swizzle)
  constexpr int LDW = K + PAD;           // LDS row stride in floats
  __shared__ float As[16 * LDW];

  const int tid  = threadIdx.x;
  const int lane = tid & 31;
  const int wave = tid >> 5;
  const int half = lane >> 4;            // 0: K pair {0,1}; 1: K pair {2,3}
  const int l16  = lane & 15;
  const int rbase = blockIdx.x * 16;
  const int cbase = wave * 16;

  // Cooperative stage of the 16xK A tile (float4 chunks, fully coalesced).
  constexpr int NCHUNK = 16 * K / 4;     // float4 elements in the tile
  for (int q = tid; q < NCHUNK; q += blockDim.x) {
    int row  = q / (K / 4);
    int col4 = q - row * (K / 4);
    v4f v = *(const v4f*)(A + (size_t)(rbase + row) * K + col4 * 4);
    *(v4f*)(&As[row * LDW + col4 * 4]) = v;
  }
  __syncthreads();

  const float* __restrict__ aptr = &As[l16 * LDW + 2 * half];
  const float* __restrict__ wptr = W + (size_t)(2 * half) * NCOL + (cbase + l16);

  v8f acc = {};
#pragma unroll 4
  for (int k = 0; k < K; k += 4) {
    v2f a = *(const v2f*)(aptr + k);               // ds_load_b64, conflict-free
    v2f b;
    b.x = wptr[(size_t)k * NCOL];
    b.y = wptr[(size_t)(k + 1) * NCOL];
    acc = __builtin_amdgcn_wmma_f32_16x16x4_f32(
        /*neg_a=*/false, a, /*neg_b=*/false, b,
        /*c_mod=*/(short)0, acc, /*reuse_a=*/false, /*reuse_b=*/false);
  }

  float* __restrict__ cptr = C + (size_t)(rbase + 8 * half) * NCOL + (cbase + l16);
#pragma unroll
  for (int r = 0; r < 8; ++r) cptr[(size_t)r * NCOL] = acc[r];
}

// ---------------------------------------------------------------------------
// Propagation: out[i,:] = dinv[i]^2 * xw[i,:] + bias  (self loop, full overwrite)
// then for each edge (s,d): out[d,:] += dinv[s]*dinv[d] * xw[s,:]
// Feature matrices (51 MB) live in the 192 MB L2, so the random gather/scatter
// runs at L2 bandwidth.
// ---------------------------------------------------------------------------
template <int NCOL>
__global__ void gcn_prop_init(const float* __restrict__ xw,
                              const float* __restrict__ dinv,
                              const float* __restrict__ bias,
                              float* __restrict__ out, int n) {
  int idx = blockIdx.x * blockDim.x + threadIdx.x;
  if (idx >= n * NCOL) return;
  int i = idx / NCOL;
  int j = idx - i * NCOL;
  float di = dinv[i];
  out[idx] = di * di * xw[idx] + bias[j];
}

template <int NCOL>
__global__ void gcn_prop_edges(const float* __restrict__ xw,
                               const int* __restrict__ src,
                               const int* __restrict__ dst,
                               const float* __restrict__ dinv,
                               float* __restrict__ out, int e) {
  const int EPB = 256 / NCOL;           // edges per 256-thread block
  int j  = threadIdx.x & (NCOL - 1);
  int el = threadIdx.x / NCOL;
  int ed = blockIdx.x * EPB + el;
  if (ed >= e) return;
  int s = src[ed], d = dst[ed];
  float w = dinv[s] * dinv[d];
  atomicAdd(&out[(size_t)d * NCOL + j], xw[(size_t)s * NCOL + j] * w);
}

// ---------------------------------------------------------------------------
// BatchNorm (training mode): column sums / sums-of-squares, then normalize+ReLU
// ---------------------------------------------------------------------------
__global__ void gcn_zero(float* __restrict__ p, int n) {
  int i = blockIdx.x * blockDim.x + threadIdx.x;
  if (i < n) p[i] = 0.0f;
}

template <int NCOL>
__global__ void gcn_bn_stats(const float* __restrict__ h,
                             float* __restrict__ sums,
                             float* __restrict__ sumsq, int n) {
  const int RL = 256 / NCOL;            // rows in flight per block
  int j  = threadIdx.x & (NCOL - 1);
  int rl = threadIdx.x / NCOL;
  float s = 0.0f, sq = 0.0f;
  for (int r = blockIdx.x * RL + rl; r < n; r += gridDim.x * RL) {
    float v = h[(size_t)r * NCOL + j];
    s += v;
    sq += v * v;
  }
  atomicAdd(&sums[j], s);
  atomicAdd(&sumsq[j], sq);
}

template <int NCOL>
__global__ void gcn_bn_relu(float* __restrict__ h,
                            const float* __restrict__ sums,
                            const float* __restrict__ sumsq,
                            const float* __restrict__ g,
                            const float* __restrict__ bt, int n) {
  int idx = blockIdx.x * blockDim.x + threadIdx.x;
  if (idx >= n * NCOL) return;
  int j = idx & (NCOL - 1);
  float inv_n = 1.0f / (float)n;
  float mu  = sums[j] * inv_n;
  float var = sumsq[j] * inv_n - mu * mu;   // biased variance == mean((h-mu)^2)
  float v = g[j] * (h[idx] - mu) * rsqrtf(var + BN_EPS) + bt[j];
  h[idx] = v > 0.0f ? v : 0.0f;
}

// ---------------------------------------------------------------------------
// Orchestration
// ---------------------------------------------------------------------------
static inline int cdiv(long long a, long long b) { return (int)((a + b - 1) / b); }

extern "C" void kernel_launch(void* const* d_in, const int* in_sizes, int n_in,
                              void* d_out, int out_size, void* d_ws, size_t ws_size,
                              hipStream_t stream) {
  const float* x   = (const float*)d_in[0];
  const int*   ei  = (const int*)d_in[1];
  const float* W1  = (const float*)d_in[2];
  const float* b1  = (const float*)d_in[3];
  const float* g1  = (const float*)d_in[4];
  const float* bt1 = (const float*)d_in[5];
  const float* W2  = (const float*)d_in[6];
  const float* b2  = (const float*)d_in[7];
  const float* g2  = (const float*)d_in[8];
  const float* bt2 = (const float*)d_in[9];
  const float* W3  = (const float*)d_in[10];
  const float* b3  = (const float*)d_in[11];
  float* out = (float*)d_out;

  const int DIN = 256, DH = 128, DOUT = 64;
  const int n = in_sizes[0] / DIN;     // 100000 (multiple of 16)
  const int E = in_sizes[1] / 2;       // 1600000
  const int* src = ei;
  const int* dst = ei + E;

  // Workspace layout (floats). n % 4 == 0 keeps everything 16B aligned.
  float* ws    = (float*)d_ws;
  float* dinv  = ws;                        // n
  float* sums  = dinv + n;                  // 128
  float* sumsq = sums + 128;                // 128
  float* bufA  = sumsq + 128;               // n*128  (GEMM outputs)
  float* bufB  = bufA + (size_t)n * DH;     // n*128  (propagated / activations)

  // --- normalization coefficients -----------------------------------------
  gcn_deg_init<<<cdiv(n, 256), 256, 0, stream>>>(dinv, n);
  gcn_deg_accum<<<cdiv(E, 256), 256, 0, stream>>>(dst, dinv, E);
  gcn_deg_to_dinv<<<cdiv(n, 256), 256, 0, stream>>>(dinv, n);

  // --- layer 1: conv(x, W1, b1) -> BN -> ReLU  ----------------------------
  gcn_gemm_wmma<256, 128><<<n / 16, 256, 0, stream>>>(x, W1, bufA);
  gcn_prop_init<128><<<cdiv((long long)n * DH, 256), 256, 0, stream>>>(bufA, dinv, b1, bufB, n);
  gcn_prop_edges<128><<<cdiv(E, 2), 256, 0, stream>>>(bufA, src, dst, dinv, bufB, E);
  gcn_zero<<<1, 256, 0, stream>>>(sums, 256);  // sums + sumsq contiguous
  gcn_bn_stats<128><<<512, 256, 0, stream>>>(bufB, sums, sumsq, n);
  gcn_bn_relu<128><<<cdiv((long long)n * DH, 256), 256, 0, stream>>>(bufB, sums, sumsq, g1, bt1, n);

  // --- layer 2: conv(h, W2, b2) -> BN -> ReLU -----------------------------
  gcn_gemm_wmma<128, 128><<<n / 16, 256, 0, stream>>>(bufB, W2, bufA);
  gcn_prop_init<128><<<cdiv((long long)n * DH, 256), 256, 0, stream>>>(bufA, dinv, b2, bufB, n);
  gcn_prop_edges<128><<<cdiv(E, 2), 256, 0, stream>>>(bufA, src, dst, dinv, bufB, E);
  gcn_zero<<<1, 256, 0, stream>>>(sums, 256);
  gcn_bn_stats<128><<<512, 256, 0, stream>>>(bufB, sums, sumsq, n);
  gcn_bn_relu<128><<<cdiv((long long)n * DH, 256), 256, 0, stream>>>(bufB, sums, sumsq, g2, bt2, n);

  // --- layer 3: conv(h, W3, b3) -> d_out ----------------------------------
  gcn_gemm_wmma<128, 64><<<n / 16, 128, 0, stream>>>(bufB, W3, bufA);
  gcn_prop_init<64><<<cdiv((long long)n * DOUT, 256), 256, 0, stream>>>(bufA, dinv, b3, out, n);
  gcn_prop_edges<64><<<cdiv(E, 4), 256, 0, stream>>>(bufA, src, dst, dinv, out, E);
}